// TopK_stablized_35270271434810
// MI455X (gfx1250) — compile-verified
//
#include <hip/hip_runtime.h>
#include <stdint.h>

#define N_COLS 8192
#define T3 256
#define EPT 32            // 8192 / 256
#define ITERS 200
#define RED_BLOCKS 512
#define RED_THREADS 256
#define TOPK_K 1024.0f
#define NU1_OVER_MU 7168.0f   // (n-K)/n / (1/n)

typedef unsigned int u32x4 __attribute__((ext_vector_type(4)));
typedef int          i32x4 __attribute__((ext_vector_type(4)));
typedef int          i32x8 __attribute__((ext_vector_type(8)));

// ---------------- stage 1: grid-stride global reductions ----------------
// Produces per-block partials: [max(s), min(s | -inf -> +inf), max over
// finite entries of max(s^2,(s-1)^2), has_neg_inf flag]
__global__ void k_red1(const float* __restrict__ x, float* __restrict__ part,
                       unsigned long long n) {
  const float NINF = -__builtin_inff();
  float mx = NINF, mn = __builtin_inff(), m2 = NINF, nf = 0.0f;
  unsigned long long stride = (unsigned long long)gridDim.x * blockDim.x;
  for (unsigned long long i =
           (unsigned long long)blockIdx.x * blockDim.x + threadIdx.x;
       i < n; i += stride) {
    float v = x[i];
    mx = fmaxf(mx, v);
    if (v == NINF) {
      nf = 1.0f;
    } else {
      mn = fminf(mn, v);
      float c = fmaxf(v * v, (v - 1.0f) * (v - 1.0f));
      m2 = fmaxf(m2, c);
    }
  }
  __shared__ float smx[RED_THREADS], smn[RED_THREADS], sm2[RED_THREADS],
      snf[RED_THREADS];
  int t = threadIdx.x;
  smx[t] = mx; smn[t] = mn; sm2[t] = m2; snf[t] = nf;
  __syncthreads();
  for (int off = RED_THREADS / 2; off > 0; off >>= 1) {
    if (t < off) {
      smx[t] = fmaxf(smx[t], smx[t + off]);
      smn[t] = fminf(smn[t], smn[t + off]);
      sm2[t] = fmaxf(sm2[t], sm2[t + off]);
      snf[t] = fmaxf(snf[t], snf[t + off]);
    }
    __syncthreads();
  }
  if (t == 0) {
    part[blockIdx.x * 4 + 0] = smx[0];
    part[blockIdx.x * 4 + 1] = smn[0];
    part[blockIdx.x * 4 + 2] = sm2[0];
    part[blockIdx.x * 4 + 3] = snf[0];
  }
}

// ---------------- stage 2: reduce the 512 partials to 4 scalars ----------------
__global__ void k_red2(const float* __restrict__ part, float* __restrict__ scal) {
  const float NINF = -__builtin_inff();
  int t = threadIdx.x;
  float mx = NINF, mn = __builtin_inff(), m2 = NINF, nf = 0.0f;
  for (int i = t; i < RED_BLOCKS; i += RED_THREADS) {
    mx = fmaxf(mx, part[i * 4 + 0]);
    mn = fminf(mn, part[i * 4 + 1]);
    m2 = fmaxf(m2, part[i * 4 + 2]);
    nf = fmaxf(nf, part[i * 4 + 3]);
  }
  __shared__ float smx[RED_THREADS], smn[RED_THREADS], sm2[RED_THREADS],
      snf[RED_THREADS];
  smx[t] = mx; smn[t] = mn; sm2[t] = m2; snf[t] = nf;
  __syncthreads();
  for (int off = RED_THREADS / 2; off > 0; off >>= 1) {
    if (t < off) {
      smx[t] = fmaxf(smx[t], smx[t + off]);
      smn[t] = fminf(smn[t], smn[t + off]);
      sm2[t] = fmaxf(sm2[t], sm2[t + off]);
      snf[t] = fmaxf(snf[t], snf[t + off]);
    }
    __syncthreads();
  }
  if (t == 0) {
    scal[0] = smx[0];
    scal[1] = smn[0];
    scal[2] = sm2[0];
    scal[3] = snf[0];
  }
}

// ---------------- stage 3: per-row exp-domain Sinkhorn ----------------
__global__ __launch_bounds__(T3) void k_sinkhorn(const float* __restrict__ scores,
                                                 float* __restrict__ out,
                                                 const float* __restrict__ scal) {
  __shared__ float tile[N_COLS];   // 32 KB row of scores, filled by TDM
  __shared__ float red[16];        // 8 waves x (S0,S1)

  const int row = blockIdx.x;
  const int t  = threadIdx.x;

  // Global scalars -> Cmax / filled value (identical on every thread)
  const float gmax = scal[0], gmin = scal[1];
  float cmax = scal[2];
  const float filled = gmin - (gmax - gmin);
  if (scal[3] > 0.5f)
    cmax = fmaxf(cmax, fmaxf(filled * filled, (filled - 1.0f) * (filled - 1.0f)));
  const float scale = __builtin_amdgcn_rcpf(0.1f * cmax);  // 1/(eps*Cmax)

  // ---- TDM: DMA this row (8192 f32 = 32KB) global -> LDS, issued by wave 0 ----
  if (t < 32) {
    uint64_t ga = (uint64_t)(uintptr_t)(scores + (size_t)row * N_COLS);
    uint32_t lds_off = (uint32_t)(uintptr_t)(void*)tile;
    // D# group0: count=1 | lds_addr | global_addr[56:0] | type=2
    u32x4 g0 = {1u, lds_off, (uint32_t)ga,
                ((uint32_t)(ga >> 32) & 0x01FFFFFFu) | 0x80000000u};
    // D# group1: data_size=4B; tensor_dim0=8192, tensor_dim1=1;
    //            tile_dim0=8192, tile_dim1=1; dim0_stride=8192
    i32x8 g1 = {0x20000,              // data_size=2 (<<16)
                (int)0x20000000,      // tensor_dim0[15:0]=8192 (<<16)
                0x10000,              // tensor_dim1=1 (<<16)
                (int)0x20000000,      // tile_dim0=8192 (<<16)
                1,                    // tile_dim1=1
                8192,                 // tensor_dim0_stride lo
                0, 0};
    i32x4 g2 = {0, 0, 0, 0};
    i32x4 g3 = {0, 0, 0, 0};
    i32x8 g4 = {0, 0, 0, 0, 0, 0, 0, 0};
    __builtin_amdgcn_tensor_load_to_lds(g0, g1, g2, g3, g4, 0);
    __builtin_amdgcn_s_wait_tensorcnt(0);
  }
  __syncthreads();

  // ---- Precompute kernel values a = exp(-s^2/(eps*Cmax)), b = exp(-(s-1)^2/..)
  float a[EPT], b[EPT];
  const float NINF = -__builtin_inff();
#pragma unroll
  for (int k = 0; k < EPT; ++k) {
    float x = tile[t + T3 * k];
    if (x == NINF) x = filled;
    a[k] = __expf(-(x * x) * scale);
    float xm1 = x - 1.0f;
    b[k] = __expf(-(xm1 * xm1) * scale);
  }

  // ---- 200 Sinkhorn iterations; state is just (v0, v1) ----
  float v0 = 1.0f, v1 = 1.0f, p0 = 1.0f, p1 = 1.0f;
  for (int it = 0; it < ITERS; ++it) {
    float r0 = 0.0f, r1 = 0.0f;
#pragma unroll
    for (int k = 0; k < EPT; ++k) {
      float d = v0 * a[k] + v1 * b[k];
      float r = __builtin_amdgcn_rcpf(d);   // u_j / mu
      r0 = fmaf(a[k], r, r0);
      r1 = fmaf(b[k], r, r1);
    }
    // wave32 butterfly reduction
#pragma unroll
    for (int off = 16; off >= 1; off >>= 1) {
      r0 += __shfl_xor(r0, off, 32);
      r1 += __shfl_xor(r1, off, 32);
    }
    __syncthreads();                 // protect red[] against previous readers
    if ((t & 31) == 0) {
      red[(t >> 5) * 2 + 0] = r0;
      red[(t >> 5) * 2 + 1] = r1;
    }
    __syncthreads();
    float S0 = 0.0f, S1 = 0.0f;
#pragma unroll
    for (int w = 0; w < 8; ++w) {
      S0 += red[2 * w + 0];
      S1 += red[2 * w + 1];
    }
    p0 = v0; p1 = v1;                // v used by this iteration's f-update
    v0 = TOPK_K / S0;                // nu0/(mu*S0)
    v1 = NU1_OVER_MU / S1;           // nu1/(mu*S1)
  }

  // ---- Output: P[0,j]*n = v0 * a_j / (p0*a_j + p1*b_j)  (Newton-refined rcp)
#pragma unroll
  for (int k = 0; k < EPT; ++k) {
    float d = p0 * a[k] + p1 * b[k];
    float r = __builtin_amdgcn_rcpf(d);
    r = r * (2.0f - d * r);
    out[(size_t)row * N_COLS + t + T3 * k] = v0 * a[k] * r;
  }
}

// ---------------- launcher ----------------
extern "C" void kernel_launch(void* const* d_in, const int* in_sizes, int n_in,
                              void* d_out, int out_size, void* d_ws, size_t ws_size,
                              hipStream_t stream) {
  const float* scores = (const float*)d_in[0];
  float* out = (float*)d_out;
  float* part = (float*)d_ws;              // RED_BLOCKS*4 floats
  float* scal = part + RED_BLOCKS * 4;     // 4 floats
  unsigned long long n = (unsigned long long)in_sizes[0];
  int rows = (int)(n / N_COLS);

  k_red1<<<RED_BLOCKS, RED_THREADS, 0, stream>>>(scores, part, n);
  k_red2<<<1, RED_THREADS, 0, stream>>>(part, scal);
  k_sinkhorn<<<rows, T3, 0, stream>>>(scores, out, scal);
}